// LocalSimGuidedSampler_14886356648189
// MI455X (gfx1250) — compile-verified
//
#include <hip/hip_runtime.h>
#include <hip/hip_bf16.h>
#include <math.h>

// ---------------------------------------------------------------------------
// LocalSimGuidedSampler for MI455X (gfx1250, wave32).
// Memory-bound pipeline (~470 MB traffic @ 23.3 TB/s ~ 20us); fused into 3
// kernels. The one real GEMM (w_dscale: 32x256 per lf pixel over 32768
// pixels) runs on V_WMMA_F32_16X16X4_F32 in exact fp32.
// Shapes: B=8, C=256, H=W=64, s=2, G=4, sH=sW=128.
// ---------------------------------------------------------------------------

typedef __attribute__((ext_vector_type(2))) float v2f;
typedef __attribute__((ext_vector_type(8))) float v8f;

#define Bn  8
#define Cn  256
#define Hn  64
#define Wn  64
#define SHn 128
#define SWn 128

// ---------------------------------------------------------------------------
// Kernel 1: hf path. Per hf pixel (yo,xo): 8 cosine sims over C=256 with the
// 8 neighbors (zero-padded borders), then
//   hr_off[j]  = b_hr_offset[j] + sum_k w_hr_offset[j,k] * sim[k]      (8x8)
//   hr_gate[j] = b_hr_dscale[j] + sum_c w_hr_dscale[j,c] * hf[c,yo,xo] (8x256)
// One block = one row (128 px) of one batch; 3-row halo staged in LDS per
// channel so hf is read ~3x (the unavoidable stencil cost).
// ---------------------------------------------------------------------------
__global__ __launch_bounds__(128)
void hf_path_kernel(const float* __restrict__ hf,
                    const float* __restrict__ w_hr_offset,
                    const float* __restrict__ b_hr_offset,
                    const float* __restrict__ w_hr_dscale,
                    const float* __restrict__ b_hr_dscale,
                    float* __restrict__ ws_hf_off,
                    float* __restrict__ ws_hf_gate) {
  const int y = blockIdx.x;        // 0..127
  const int b = blockIdx.y;        // 0..7
  const int x = threadIdx.x;       // 0..127

  __shared__ float srow[3][SWn + 2];

  float cacc = 0.f;
  float dot[8], nn[8], gacc[8];
#pragma unroll
  for (int k = 0; k < 8; ++k) { dot[k] = 0.f; nn[k] = 0.f; gacc[k] = 0.f; }

  const float* base = hf + (size_t)b * Cn * SHn * SWn;

  for (int c = 0; c < Cn; ++c) {
    const float* plane = base + (size_t)c * SHn * SWn;
#pragma unroll
    for (int r = 0; r < 3; ++r) {
      const int yy = y + r - 1;
      for (int i = threadIdx.x; i < SWn + 2; i += 128) {
        const int xx = i - 1;
        float v = 0.f;
        if (yy >= 0 && yy < SHn && xx >= 0 && xx < SWn) v = plane[yy * SWn + xx];
        srow[r][i] = v;
      }
    }
    __syncthreads();
    const float v = srow[1][x + 1];
    cacc = fmaf(v, v, cacc);
    int k = 0;
#pragma unroll
    for (int r = 0; r < 3; ++r) {
#pragma unroll
      for (int dx = -1; dx <= 1; ++dx) {
        if (r == 1 && dx == 0) continue;
        const float nb = srow[r][x + 1 + dx];
        dot[k] = fmaf(v, nb, dot[k]);
        nn[k]  = fmaf(nb, nb, nn[k]);
        ++k;
      }
    }
#pragma unroll
    for (int j = 0; j < 8; ++j)
      gacc[j] = fmaf(w_hr_dscale[j * Cn + c], v, gacc[j]);  // uniform -> s_load
    __syncthreads();
  }

  float sims[8];
  const float cn = sqrtf(cacc);
#pragma unroll
  for (int k = 0; k < 8; ++k) {
    const float denom = fmaxf(cn * sqrtf(nn[k]), 1e-8f);
    sims[k] = dot[k] / denom;
  }

  const size_t pix = (size_t)y * SWn + x;
#pragma unroll
  for (int j = 0; j < 8; ++j) {
    float o = b_hr_offset[j];
#pragma unroll
    for (int k = 0; k < 8; ++k) o = fmaf(w_hr_offset[j * 8 + k], sims[k], o);
    const size_t idx = ((size_t)(b * 8 + j) * SHn * SWn) + pix;
    ws_hf_off[idx]  = o;
    ws_hf_gate[idx] = gacc[j] + b_hr_dscale[j];
  }
}

// ---------------------------------------------------------------------------
// Kernel 2: lf path. One block = one lf row (64 px) of one batch, 64 threads
// (2 waves).
//  Phase A (VALU): 8 cosine sims -> lf_off[32] = w_offset(32x8) @ sims + b.
//  Phase B (WMMA f32 16x16x4): lf_gate_raw[32] = w_dscale(32x256) @ lf + b.
//    Each wave owns 32 pixels = 2 N-tiles; M=32 = 2 M-tiles; K walks 256 in
//    steps of 4 -> 4 x 64 = 256 v_wmma_f32_16x16x4_f32 per wave.
//    A frag: lane L holds w_dscale[M = (L&15)+16*mt][K = k0+2*(L>>4)+j]
//    B frag: lane L holds lf  [K = k0+2*(L>>4)+j][pixel = 16*nt+(L&15)]
//    D:      lane L, reg r -> M = 16*mt + r + 8*(L>>4), pixel = 16*nt+(L&15)
// ---------------------------------------------------------------------------
__global__ __launch_bounds__(64)
void lf_path_kernel(const float* __restrict__ lf,
                    const float* __restrict__ w_offset,
                    const float* __restrict__ b_offset,
                    const float* __restrict__ w_dscale,
                    const float* __restrict__ b_dscale,
                    float* __restrict__ ws_lf_off,
                    float* __restrict__ ws_lf_gate) {
  const int h = blockIdx.x;        // 0..63
  const int b = blockIdx.y;        // 0..7
  const int x = threadIdx.x;       // 0..63

  __shared__ float srow[3][Wn + 2];

  // ---- Phase A: direction sims + 32x8 offset conv -------------------------
  float cacc = 0.f;
  float dot[8], nn[8];
#pragma unroll
  for (int k = 0; k < 8; ++k) { dot[k] = 0.f; nn[k] = 0.f; }

  const float* base = lf + (size_t)b * Cn * Hn * Wn;
  for (int c = 0; c < Cn; ++c) {
    const float* plane = base + (size_t)c * Hn * Wn;
#pragma unroll
    for (int r = 0; r < 3; ++r) {
      const int yy = h + r - 1;
      for (int i = threadIdx.x; i < Wn + 2; i += 64) {
        const int xx = i - 1;
        float v = 0.f;
        if (yy >= 0 && yy < Hn && xx >= 0 && xx < Wn) v = plane[yy * Wn + xx];
        srow[r][i] = v;
      }
    }
    __syncthreads();
    const float v = srow[1][x + 1];
    cacc = fmaf(v, v, cacc);
    int k = 0;
#pragma unroll
    for (int r = 0; r < 3; ++r) {
#pragma unroll
      for (int dx = -1; dx <= 1; ++dx) {
        if (r == 1 && dx == 0) continue;
        const float nb = srow[r][x + 1 + dx];
        dot[k] = fmaf(v, nb, dot[k]);
        nn[k]  = fmaf(nb, nb, nn[k]);
        ++k;
      }
    }
    __syncthreads();
  }

  float sims[8];
  const float cn = sqrtf(cacc);
#pragma unroll
  for (int k = 0; k < 8; ++k) {
    const float denom = fmaxf(cn * sqrtf(nn[k]), 1e-8f);
    sims[k] = dot[k] / denom;
  }

  {
    const size_t obase = (size_t)b * 32 * Hn * Wn + (size_t)h * Wn + x;
    for (int m = 0; m < 32; ++m) {
      float o = b_offset[m];
#pragma unroll
      for (int k = 0; k < 8; ++k) o = fmaf(w_offset[m * 8 + k], sims[k], o);
      ws_lf_off[obase + (size_t)m * Hn * Wn] = o;
    }
  }

  // ---- Phase B: 32x256 gate conv via V_WMMA_F32_16X16X4_F32 ---------------
  const int lane = threadIdx.x & 31;
  const int wv   = threadIdx.x >> 5;   // wave id: pixels [32*wv, 32*wv+32)
  const int lo   = lane & 15;
  const int hi   = lane >> 4;

  v8f acc00 = {}; v8f acc01 = {}; v8f acc10 = {}; v8f acc11 = {};
  const float* lfrow = lf + (size_t)b * Cn * Hn * Wn + (size_t)h * Wn;
  const int xp0 = wv * 32 + lo;
  const int xp1 = xp0 + 16;

  for (int k0 = 0; k0 < Cn; k0 += 4) {
    const int ka = k0 + 2 * hi;
    v2f a0, a1, b0v, b1v;
    a0.x = w_dscale[lo * Cn + ka];
    a0.y = w_dscale[lo * Cn + ka + 1];
    a1.x = w_dscale[(lo + 16) * Cn + ka];
    a1.y = w_dscale[(lo + 16) * Cn + ka + 1];
    b0v.x = lfrow[(size_t)ka * Hn * Wn + xp0];
    b0v.y = lfrow[(size_t)(ka + 1) * Hn * Wn + xp0];
    b1v.x = lfrow[(size_t)ka * Hn * Wn + xp1];
    b1v.y = lfrow[(size_t)(ka + 1) * Hn * Wn + xp1];
    acc00 = __builtin_amdgcn_wmma_f32_16x16x4_f32(false, a0, false, b0v, (short)0, acc00, false, false);
    acc01 = __builtin_amdgcn_wmma_f32_16x16x4_f32(false, a0, false, b1v, (short)0, acc01, false, false);
    acc10 = __builtin_amdgcn_wmma_f32_16x16x4_f32(false, a1, false, b0v, (short)0, acc10, false, false);
    acc11 = __builtin_amdgcn_wmma_f32_16x16x4_f32(false, a1, false, b1v, (short)0, acc11, false, false);
  }

  float* gout = ws_lf_gate + (size_t)b * 32 * Hn * Wn + (size_t)h * Wn;
#pragma unroll
  for (int r = 0; r < 8; ++r) {
    const int M0 = r + 8 * hi;
    const int M1 = 16 + r + 8 * hi;
    gout[(size_t)M0 * Hn * Wn + xp0] = acc00[r] + b_dscale[M0];
    gout[(size_t)M0 * Hn * Wn + xp1] = acc01[r] + b_dscale[M0];
    gout[(size_t)M1 * Hn * Wn + xp0] = acc10[r] + b_dscale[M1];
    gout[(size_t)M1 * Hn * Wn + xp1] = acc11[r] + b_dscale[M1];
  }
}

// ---------------------------------------------------------------------------
// Kernel 3: assemble offsets, sigmoid gate, bilinear border sample.
// Un-normalized grid math reduces to: sample feat group-g plane at
// (w + offx, h + offy) clamped to [0, 63]. One thread per output column,
// loop over the 64 group channels (1 MB slab -> L2 resident).
// ---------------------------------------------------------------------------
__global__ __launch_bounds__(128)
void sample_kernel(const float* __restrict__ feat,
                   const float* __restrict__ ws_lf_off,
                   const float* __restrict__ ws_lf_gate,
                   const float* __restrict__ ws_hf_off,
                   const float* __restrict__ ws_hf_gate,
                   float* __restrict__ out) {
  const int yo = blockIdx.x;       // 0..127
  const int g  = blockIdx.y;       // 0..3
  const int b  = blockIdx.z;       // 0..7
  const int xo = threadIdx.x;      // 0..127

  const int h = yo >> 1, a = yo & 1;
  const int w = xo >> 1, bb = xo & 1;
  const int k = g * 4 + a * 2 + bb;

  const size_t lfpix  = (size_t)h * Wn + w;
  const size_t lfbase = (size_t)b * 32 * Hn * Wn;
  const float offx_p = ws_lf_off [lfbase + (size_t)k        * Hn * Wn + lfpix];
  const float offy_p = ws_lf_off [lfbase + (size_t)(16 + k) * Hn * Wn + lfpix];
  const float gx_p   = ws_lf_gate[lfbase + (size_t)k        * Hn * Wn + lfpix];
  const float gy_p   = ws_lf_gate[lfbase + (size_t)(16 + k) * Hn * Wn + lfpix];

  const size_t hfpix  = (size_t)yo * SWn + xo;
  const size_t hfbase = (size_t)b * 8 * SHn * SWn;
  const float offx_h = ws_hf_off [hfbase + (size_t)g       * SHn * SWn + hfpix];
  const float offy_h = ws_hf_off [hfbase + (size_t)(4 + g) * SHn * SWn + hfpix];
  const float gx_h   = ws_hf_gate[hfbase + (size_t)g       * SHn * SWn + hfpix];
  const float gy_h   = ws_hf_gate[hfbase + (size_t)(4 + g) * SHn * SWn + hfpix];

  const float gatex = 1.f / (1.f + expf(-(gx_p + gx_h)));
  const float gatey = 1.f / (1.f + expf(-(gy_p + gy_h)));
  const float offx = (offx_p + offx_h) * gatex + (bb ? 0.25f : -0.25f);
  const float offy = (offy_p + offy_h) * gatey + (a  ? 0.25f : -0.25f);

  const float gx = fminf(fmaxf((float)w + offx, 0.f), (float)(Wn - 1));
  const float gy = fminf(fmaxf((float)h + offy, 0.f), (float)(Hn - 1));
  const float x0f = floorf(gx), y0f = floorf(gy);
  const float wx = gx - x0f,  wy = gy - y0f;
  const int x0 = min(max((int)x0f, 0), Wn - 1);
  const int y0 = min(max((int)y0f, 0), Hn - 1);
  const int x1 = min(x0 + 1, Wn - 1);
  const int y1 = min(y0 + 1, Hn - 1);
  const float w00 = (1.f - wx) * (1.f - wy);
  const float w01 = wx * (1.f - wy);
  const float w10 = (1.f - wx) * wy;
  const float w11 = wx * wy;
  const int i00 = y0 * Wn + x0, i01 = y0 * Wn + x1;
  const int i10 = y1 * Wn + x0, i11 = y1 * Wn + x1;

  const float* fb = feat + ((size_t)b * Cn + g * 64) * Hn * Wn;
  float* ob = out + (((size_t)b * Cn + g * 64) * SHn + yo) * SWn + xo;
#pragma unroll 4
  for (int c = 0; c < 64; ++c) {
    const float* p = fb + (size_t)c * Hn * Wn;
    const float v = p[i00] * w00 + p[i01] * w01 + p[i10] * w10 + p[i11] * w11;
    ob[(size_t)c * SHn * SWn] = v;
  }
}

// ---------------------------------------------------------------------------
extern "C" void kernel_launch(void* const* d_in, const int* in_sizes, int n_in,
                              void* d_out, int out_size, void* d_ws, size_t ws_size,
                              hipStream_t stream) {
  (void)in_sizes; (void)n_in; (void)out_size; (void)ws_size;
  const float* hf          = (const float*)d_in[0];
  const float* lf          = (const float*)d_in[1];
  const float* feat        = (const float*)d_in[2];
  const float* w_offset    = (const float*)d_in[3];
  const float* b_offset    = (const float*)d_in[4];
  const float* w_dscale    = (const float*)d_in[5];
  const float* b_dscale    = (const float*)d_in[6];
  const float* w_hr_offset = (const float*)d_in[7];
  const float* b_hr_offset = (const float*)d_in[8];
  const float* w_hr_dscale = (const float*)d_in[9];
  const float* b_hr_dscale = (const float*)d_in[10];
  float* out = (float*)d_out;
  float* ws  = (float*)d_ws;

  const size_t HFN = (size_t)Bn * 8  * SHn * SWn;  // 1,048,576 floats
  const size_t LFN = (size_t)Bn * 32 * Hn  * Wn;   // 1,048,576 floats
  float* ws_hf_off  = ws;
  float* ws_hf_gate = ws + HFN;
  float* ws_lf_off  = ws + 2 * HFN;
  float* ws_lf_gate = ws + 2 * HFN + LFN;

  hf_path_kernel<<<dim3(SHn, Bn), 128, 0, stream>>>(
      hf, w_hr_offset, b_hr_offset, w_hr_dscale, b_hr_dscale,
      ws_hf_off, ws_hf_gate);

  lf_path_kernel<<<dim3(Hn, Bn), 64, 0, stream>>>(
      lf, w_offset, b_offset, w_dscale, b_dscale,
      ws_lf_off, ws_lf_gate);

  sample_kernel<<<dim3(SHn, 4, Bn), 128, 0, stream>>>(
      feat, ws_lf_off, ws_lf_gate, ws_hf_off, ws_hf_gate, out);
}